// AMICOSolver_30726196035758
// MI455X (gfx1250) — compile-verified
//
#include <hip/hip_runtime.h>

// ---------------------------------------------------------------------------
// ADMM nonnegative LASSO:  min ||Y^T - A x||^2 + lambda ||x||_1, x >= 0
//   Y: [4096 x 256] f32, A: [256 x 512] f32, out X_hat: [4096 x 512] f32
// Plan:
//   At  = A^T                  [512 x 256]      (transpose kernel)
//   Yt  = Y^T                  [256 x 4096]     (transpose kernel)
//   AtY = At @ Yt              [512 x 4096]     (WMMA f32 GEMM)
//   G   = At @ A + I           [512 x 512]      (WMMA f32 GEMM + diag add)
//   Minv= G^{-1} via Newton-Schulz X <- X(2I - G X), X0 = I/||G||inf
//                                                   (2 WMMA GEMMs / NS iter)
//   30x ADMM:  x = Minv @ rhs  (WMMA GEMM),  prox/elementwise updates
//   out = x^T
// All GEMMs use V_WMMA_F32_16X16X4_F32 (full fp32 matrix pipe on CDNA5).
// ---------------------------------------------------------------------------

typedef float v2f __attribute__((ext_vector_type(2)));
typedef float v8f __attribute__((ext_vector_type(8)));

#define B_VOX   4096
#define M_MEAS  256
#define K_ATOMS 512
#define ADMM_ITERS 30
#define NS_ITERS   22
#define LAMBDA_REG 0.1f

// ---------------------------------------------------------------------------
// Generic row-major f32 GEMM: C[M x N] = A[M x K] @ B[K x N]
// Wave tile: 32(M) x 64(N); 8 waves / block; requires M%32==0, N%64==0, K%4==0
// ---------------------------------------------------------------------------
__global__ __launch_bounds__(256)
void gemm_wmma_f32(const float* __restrict__ A, const float* __restrict__ B,
                   float* __restrict__ C, int M, int N, int K) {
  const int lane  = threadIdx.x & 31;
  const int wave  = threadIdx.x >> 5;
  const int mtiles = M >> 5;   // 32-row tiles
  const int ntiles = N >> 6;   // 64-col tiles
  const int t = blockIdx.x * 8 + wave;
  if (t >= mtiles * ntiles) return;          // uniform per wave

  const int m0 = (t % mtiles) << 5;
  const int n0 = (t / mtiles) << 6;
  const int lm = lane & 15;                  // row/col within 16-tile
  const int lk = (lane >> 4) << 1;           // K sub-offset: 0 or 2

  v8f acc[2][4] = {};                        // 2(M) x 4(N) accumulators

  const float* Ap0 = A + (size_t)(m0 + lm) * K + lk;
  const float* Ap1 = Ap0 + (size_t)16 * K;
  const float* Bp  = B + (size_t)lk * N + n0 + lm;

#pragma unroll 2
  for (int k = 0; k < K; k += 4) {
    // A fragments (16x4 f32): lane -> row lm, VGPR pair -> K = k+lk, k+lk+1
    v2f a0 = *(const v2f*)(Ap0 + k);
    v2f a1 = *(const v2f*)(Ap1 + k);
    // B fragments (4x16 f32): lane -> col lm, VGPR pair -> K = k+lk, k+lk+1
    const float* bp = Bp + (size_t)k * N;
    v2f b0, b1, b2, b3;
    b0.x = bp[0];       b0.y = bp[N];
    b1.x = bp[16];      b1.y = bp[16 + N];
    b2.x = bp[32];      b2.y = bp[32 + N];
    b3.x = bp[48];      b3.y = bp[48 + N];

    acc[0][0] = __builtin_amdgcn_wmma_f32_16x16x4_f32(false, a0, false, b0, (short)0, acc[0][0], false, false);
    acc[0][1] = __builtin_amdgcn_wmma_f32_16x16x4_f32(false, a0, false, b1, (short)0, acc[0][1], false, false);
    acc[0][2] = __builtin_amdgcn_wmma_f32_16x16x4_f32(false, a0, false, b2, (short)0, acc[0][2], false, false);
    acc[0][3] = __builtin_amdgcn_wmma_f32_16x16x4_f32(false, a0, false, b3, (short)0, acc[0][3], false, false);
    acc[1][0] = __builtin_amdgcn_wmma_f32_16x16x4_f32(false, a1, false, b0, (short)0, acc[1][0], false, false);
    acc[1][1] = __builtin_amdgcn_wmma_f32_16x16x4_f32(false, a1, false, b1, (short)0, acc[1][1], false, false);
    acc[1][2] = __builtin_amdgcn_wmma_f32_16x16x4_f32(false, a1, false, b2, (short)0, acc[1][2], false, false);
    acc[1][3] = __builtin_amdgcn_wmma_f32_16x16x4_f32(false, a1, false, b3, (short)0, acc[1][3], false, false);
  }

  // C/D layout: lane l, vgpr r -> row (r + 8*(l/16)), col l%16 of the 16x16 tile
  const int mrow = m0 + ((lane >> 4) << 3);
  float* Cp = C + (size_t)mrow * N + n0 + lm;
#pragma unroll
  for (int i = 0; i < 2; ++i)
#pragma unroll
    for (int j = 0; j < 4; ++j)
#pragma unroll
      for (int r = 0; r < 8; ++r)
        Cp[(size_t)(16 * i + r) * N + 16 * j] = acc[i][j][r];
}

// ---------------------------------------------------------------------------
// Helper kernels
// ---------------------------------------------------------------------------
__global__ void transpose_f32(const float* __restrict__ in, float* __restrict__ out,
                              int R, int C) {   // in: RxC -> out: CxR (coalesced writes)
  long o = (long)blockIdx.x * blockDim.x + threadIdx.x;
  long total = (long)R * C;
  if (o >= total) return;
  int  r = (int)(o % R);
  long c = o / R;
  out[o] = in[(long)r * C + c];
}

__global__ void add_diag_f32(float* __restrict__ G, int n, float v) {
  int i = blockIdx.x * blockDim.x + threadIdx.x;
  if (i < n) G[(size_t)i * n + i] += v;
}

__global__ void row_abs_sum_f32(const float* __restrict__ G, float* __restrict__ rs, int n) {
  int i = blockIdx.x * blockDim.x + threadIdx.x;
  if (i >= n) return;
  float s = 0.0f;
  for (int j = 0; j < n; ++j) s += fabsf(G[(size_t)i * n + j]);
  rs[i] = s;
}

__global__ void reduce_inv_max_f32(const float* __restrict__ rs, float* __restrict__ alpha) {
  __shared__ float sm[256];
  int t = threadIdx.x;
  float m = fmaxf(rs[t], rs[t + 256]);       // n == 512
  sm[t] = m;
  __syncthreads();
  for (int s = 128; s > 0; s >>= 1) {
    if (t < s) sm[t] = fmaxf(sm[t], sm[t + s]);
    __syncthreads();
  }
  if (t == 0) alpha[0] = 1.0f / sm[0];
}

__global__ void init_scaled_identity_f32(float* __restrict__ X, const float* __restrict__ alpha, int n) {
  long idx = (long)blockIdx.x * blockDim.x + threadIdx.x;
  if (idx >= (long)n * n) return;
  int r = (int)(idx / n), c = (int)(idx % n);
  X[idx] = (r == c) ? alpha[0] : 0.0f;
}

__global__ void two_i_minus_inplace_f32(float* __restrict__ T, int n) {
  long idx = (long)blockIdx.x * blockDim.x + threadIdx.x;
  if (idx >= (long)n * n) return;
  int r = (int)(idx / n), c = (int)(idx % n);
  T[idx] = ((r == c) ? 2.0f : 0.0f) - T[idx];
}

__global__ void zero_f32(float* __restrict__ p, long n) {
  long idx = (long)blockIdx.x * blockDim.x + threadIdx.x;
  if (idx < n) p[idx] = 0.0f;
}

// z = max(x+u - lambda, 0); u' = u + x - z; rhs = AtY + z - u'  (rho == 1)
__global__ void prox_update_f32(const float* __restrict__ x, float* __restrict__ u,
                                float* __restrict__ rhs, const float* __restrict__ AtY,
                                long n) {
  long idx = (long)blockIdx.x * blockDim.x + threadIdx.x;
  if (idx >= n) return;
  float xi = x[idx], ui = u[idx];
  float v  = xi + ui;
  float z  = fmaxf(v - LAMBDA_REG, 0.0f);    // == soft-threshold(max(v,0))
  float un = ui + xi - z;
  u[idx]   = un;
  rhs[idx] = AtY[idx] + z - un;
}

// ---------------------------------------------------------------------------
static inline int gemm_blocks(int M, int N) { return ((M >> 5) * (N >> 6) + 7) / 8; }

extern "C" void kernel_launch(void* const* d_in, const int* in_sizes, int n_in,
                              void* d_out, int out_size, void* d_ws, size_t ws_size,
                              hipStream_t stream) {
  const float* Y = (const float*)d_in[0];   // [4096 x 256]
  const float* A = (const float*)d_in[1];   // [256 x 512]
  float* out = (float*)d_out;               // [4096 x 512]
  (void)in_sizes; (void)n_in; (void)out_size; (void)ws_size;

  const int n  = K_ATOMS;                   // 512
  const long NB = (long)K_ATOMS * B_VOX;    // 512*4096 elements

  // Workspace layout (floats); Yt aliases rhs, At aliases x (setup-only uses)
  float* f     = (float*)d_ws;
  float* G     = f;                         // 512*512
  float* T     = G   + (size_t)n * n;       // 512*512
  float* X0    = T   + (size_t)n * n;       // 512*512
  float* X1    = X0  + (size_t)n * n;       // 512*512
  float* AtY   = X1  + (size_t)n * n;       // 512*4096
  float* rhs   = AtY + NB;                  // 512*4096  (alias: Yt 256*4096)
  float* x     = rhs + NB;                  // 512*4096  (alias: At 512*256)
  float* u     = x   + NB;                  // 512*4096
  float* rsum  = u   + NB;                  // 512
  float* alpha = rsum + n;                  // 1
  float* Yt = rhs;
  float* At = x;

  // --- Setup: transposes ---
  transpose_f32<<<((long)M_MEAS * K_ATOMS + 255) / 256, 256, 0, stream>>>(A, At, M_MEAS, K_ATOMS);
  transpose_f32<<<((long)B_VOX * M_MEAS + 255) / 256, 256, 0, stream>>>(Y, Yt, B_VOX, M_MEAS);

  // AtY = At @ Yt   [512 x 4096], K = 256
  gemm_wmma_f32<<<gemm_blocks(K_ATOMS, B_VOX), 256, 0, stream>>>(At, Yt, AtY, K_ATOMS, B_VOX, M_MEAS);

  // G = At @ A + I  [512 x 512], K = 256 (rho == 1)
  gemm_wmma_f32<<<gemm_blocks(n, n), 256, 0, stream>>>(At, A, G, n, n, M_MEAS);
  add_diag_f32<<<2, 256, 0, stream>>>(G, n, 1.0f);

  // alpha = 1 / ||G||_inf ; X0 = alpha * I
  row_abs_sum_f32<<<2, 256, 0, stream>>>(G, rsum, n);
  reduce_inv_max_f32<<<1, 256, 0, stream>>>(rsum, alpha);
  init_scaled_identity_f32<<<((long)n * n + 255) / 256, 256, 0, stream>>>(X0, alpha, n);

  // --- Newton-Schulz inversion: X <- X (2I - G X) ---
  float* Xc = X0; float* Xn = X1;
  for (int it = 0; it < NS_ITERS; ++it) {
    gemm_wmma_f32<<<gemm_blocks(n, n), 256, 0, stream>>>(G, Xc, T, n, n, n);
    two_i_minus_inplace_f32<<<((long)n * n + 255) / 256, 256, 0, stream>>>(T, n);
    gemm_wmma_f32<<<gemm_blocks(n, n), 256, 0, stream>>>(Xc, T, Xn, n, n, n);
    float* tmp = Xc; Xc = Xn; Xn = tmp;
  }
  const float* Minv = Xc;

  // --- ADMM main loop (z0 = u0 = 0 => first rhs = AtY) ---
  zero_f32<<<(NB + 255) / 256, 256, 0, stream>>>(u, NB);
  for (int it = 0; it < ADMM_ITERS; ++it) {
    const float* Bop = (it == 0) ? AtY : rhs;
    gemm_wmma_f32<<<gemm_blocks(K_ATOMS, B_VOX), 256, 0, stream>>>(Minv, Bop, x, K_ATOMS, B_VOX, K_ATOMS);
    prox_update_f32<<<(NB + 255) / 256, 256, 0, stream>>>(x, u, rhs, AtY, NB);
  }

  // out = x^T   [4096 x 512]
  transpose_f32<<<(NB + 255) / 256, 256, 0, stream>>>(x, out, K_ATOMS, B_VOX);
}